// YOLOPOST_36137854828808
// MI455X (gfx1250) — compile-verified
//
#include <hip/hip_runtime.h>
#include <stdint.h>

typedef __attribute__((ext_vector_type(4))) float floatx4;
typedef __attribute__((ext_vector_type(4))) int   intx4;

// ---------------------------------------------------------------------------
// Counter-based RNG (deterministic per element index per call, no state)
// ---------------------------------------------------------------------------
__device__ __forceinline__ unsigned pcg(unsigned x) {
    x = x * 747796405u + 2891336453u;
    unsigned w = ((x >> ((x >> 28) + 4u)) ^ x) * 277803737u;
    return (w >> 22) ^ w;
}
__device__ __forceinline__ float u01(unsigned h) {
    return (float)(h >> 8) * (1.0f / 16777216.0f); // [0,1)
}
// Box-Muller: two N(0,1) samples from one (log,sqrt,sin,cos) set
__device__ __forceinline__ void gauss2(unsigned id, float& g0, float& g1) {
    unsigned h1 = pcg(id ^ 0x68bc21ebu);
    unsigned h2 = pcg(id + 0x02e5be93u);
    float u1 = u01(h1) + 5.9604645e-8f;   // avoid log(0)
    float a  = 6.2831853f * u01(h2);
    float r  = __fsqrt_rn(-2.0f * __logf(u1));
    g0 = r * __cosf(a);
    g1 = r * __sinf(a);
}

// ---------------------------------------------------------------------------
// Stage 0: reset the three float-min slots to +INF (ws is poisoned / stale)
// ---------------------------------------------------------------------------
__global__ void init_min(float* mins) {
    if (threadIdx.x < 3) mins[threadIdx.x] = __int_as_float(0x7F800000); // +INF
}

// ---------------------------------------------------------------------------
// Stage 1: per-position argmax over the 80 class channels of batch 0,
//          plus global min of that slab.
// CDNA5 path: async DMA of the strided [80 x 160] tile into LDS
// (global_load_async_to_lds_b128, ASYNCcnt), then conflict-free LDS columns.
// LDS: 80*160*4 = 50 KB / workgroup (320 KB per WGP available).
// ---------------------------------------------------------------------------
template <int N>
__global__ __launch_bounds__(160) void argmax_min_kernel(
    const float* __restrict__ x, int* __restrict__ amax, float* __restrict__ minOut)
{
    __shared__ __align__(16) float tile[80 * 160];
    __shared__ float wmin[5];

    const int tid = threadIdx.x;
    const int n0  = blockIdx.x * 160;

    // batch 0, channel-5 row start for this column block
    const float* base = x + (size_t)5 * N + n0;

    // 80 rows x 160 floats = 3200 float4s; 20 async b128 loads per thread
    #pragma unroll
    for (int k = 0; k < 20; ++k) {
        int f4 = tid + k * 160;          // float4 index in tile
        int c  = f4 / 40;                // class channel 0..79
        int j4 = f4 % 40;                // float4 within row
        const float* gp = base + (size_t)c * N + (size_t)j4 * 4;
        unsigned lds_addr = (unsigned)(uintptr_t)(&tile[f4 * 4]);
        asm volatile("global_load_async_to_lds_b128 %0, %1, off"
                     :: "v"(lds_addr), "v"((unsigned long long)(uintptr_t)gp)
                     : "memory");
    }
    asm volatile("s_wait_asynccnt 0" ::: "memory");
    __syncthreads();

    // Column argmax (first-max tie-break, matches jnp.argmax) + column min.
    float best = tile[tid];
    float mn   = best;
    int   bi   = 0;
    #pragma unroll 4
    for (int c = 1; c < 80; ++c) {
        float v = tile[c * 160 + tid];
        if (v > best) { best = v; bi = c; }
        mn = fminf(mn, v);
    }
    amax[n0 + tid] = bi;

    // wave32 reduce, then 5-wave block reduce, then sign-aware atomic min
    #pragma unroll
    for (int off = 16; off > 0; off >>= 1)
        mn = fminf(mn, __shfl_xor(mn, off, 32));
    if ((tid & 31) == 0) wmin[tid >> 5] = mn;
    __syncthreads();
    if (tid == 0) {
        float m = wmin[0];
        #pragma unroll
        for (int i = 1; i < 5; ++i) m = fminf(m, wmin[i]);
        if (m >= 0.0f) atomicMin((int*)minOut, __float_as_int(m));
        else           atomicMax((unsigned int*)minOut, __float_as_uint(m));
    }
}

// ---------------------------------------------------------------------------
// Stage 2: streaming float4 pass — replace non-argmax batch-0 class logits
// with uniform()*min, then add N(0, value) noise to everything.
// Last-use NT loads for x, NT stores for out (CDNA5 TH cache hints) so the
// 192MB L2 keeps only the reused argmax/min/value data.
// ---------------------------------------------------------------------------
template <int N>
__global__ __launch_bounds__(256) void yolo_noise_kernel(
    const float* __restrict__ x, float* __restrict__ out,
    const int* __restrict__ amax, const float* __restrict__ minPtr,
    const float* __restrict__ valuePtr, size_t total4, unsigned gbase)
{
    size_t idx4 = (size_t)blockIdx.x * 256 + threadIdx.x;
    if (idx4 >= total4) return;

    const float value = *valuePtr;
    const float mn    = *minPtr;

    size_t e  = idx4 * 4;                 // N % 4 == 0, so e..e+3 share (b,c)
    int    n  = (int)(e % (size_t)N);
    int    bc = (int)(e / (size_t)N);
    int    c  = bc % 85;
    int    b  = bc / 85;

    floatx4 v4 = __builtin_nontemporal_load(reinterpret_cast<const floatx4*>(x) + idx4);
    float v[4] = {v4.x, v4.y, v4.z, v4.w};
    unsigned gid0 = gbase + (unsigned)e;

    if (b == 0 && c >= 5) {
        intx4 a4 = reinterpret_cast<const intx4*>(amax)[n >> 2];
        int   a[4] = {a4.x, a4.y, a4.z, a4.w};
        int   k = c - 5;
        #pragma unroll
        for (int j = 0; j < 4; ++j)
            if (a[j] != k)
                v[j] = u01(pcg((gid0 + j) * 2654435761u + 1013904223u)) * mn;
    }

    float g[4];
    gauss2(gid0,      g[0], g[1]);
    gauss2(gid0 + 2u, g[2], g[3]);
    #pragma unroll
    for (int j = 0; j < 4; ++j)
        v[j] += g[j] * value;

    floatx4 r = {v[0], v[1], v[2], v[3]};
    __builtin_nontemporal_store(r, reinterpret_cast<floatx4*>(out) + idx4);
}

// ---------------------------------------------------------------------------
// Launch
// ---------------------------------------------------------------------------
extern "C" void kernel_launch(void* const* d_in, const int* in_sizes, int n_in,
                              void* d_out, int out_size, void* d_ws, size_t ws_size,
                              hipStream_t stream) {
    (void)in_sizes; (void)n_in; (void)out_size; (void)ws_size;

    const float* xs[3]  = {(const float*)d_in[0], (const float*)d_in[1], (const float*)d_in[2]};
    const float* value  = (const float*)d_in[3];
    float*       out    = (float*)d_out;

    constexpr int N0 = 160 * 160, N1 = 80 * 80, N2 = 40 * 40;

    float* mins = (float*)d_ws;                 // 3 floats (16B-aligned base)
    int*   am   = ((int*)d_ws) + 4;             // argmax tables, 33600 ints total
    int*   am_t[3]  = {am, am + N0, am + N0 + N1};
    size_t off_t[3] = {0, (size_t)8 * 85 * N0, (size_t)8 * 85 * (N0 + N1)};

    init_min<<<1, 32, 0, stream>>>(mins);

    argmax_min_kernel<N0><<<N0 / 160, 160, 0, stream>>>(xs[0], am_t[0], mins + 0);
    argmax_min_kernel<N1><<<N1 / 160, 160, 0, stream>>>(xs[1], am_t[1], mins + 1);
    argmax_min_kernel<N2><<<N2 / 160, 160, 0, stream>>>(xs[2], am_t[2], mins + 2);

    const size_t t4_0 = (size_t)170 * N0;  // 8*85*N/4
    const size_t t4_1 = (size_t)170 * N1;
    const size_t t4_2 = (size_t)170 * N2;
    yolo_noise_kernel<N0><<<(int)((t4_0 + 255) / 256), 256, 0, stream>>>(
        xs[0], out + off_t[0], am_t[0], mins + 0, value, t4_0, (unsigned)off_t[0]);
    yolo_noise_kernel<N1><<<(int)((t4_1 + 255) / 256), 256, 0, stream>>>(
        xs[1], out + off_t[1], am_t[1], mins + 1, value, t4_1, (unsigned)off_t[1]);
    yolo_noise_kernel<N2><<<(int)((t4_2 + 255) / 256), 256, 0, stream>>>(
        xs[2], out + off_t[2], am_t[2], mins + 2, value, t4_2, (unsigned)off_t[2]);
}